// GraphConv_47038481825892
// MI455X (gfx1250) — compile-verified
//
#include <hip/hip_runtime.h>

// Problem constants (match reference)
#define N_NODES  65536
#define C_IN     128
#define C_OUT    128
#define N_ETYPE  7
#define N_NTYPE  5
#define FEAT     133            // C_IN + N_NTYPE
#define K_REAL   931            // 7 * 133
#define KP       960            // padded K, multiple of 64
#define SLOT6    798            // 6 * 133 : self-loop slot start

typedef __attribute__((ext_vector_type(2))) float v2f;
typedef __attribute__((ext_vector_type(8))) float v8f;

// ---------------- workspace layout (bytes) ----------------
// col_data : [N][KP] f32                       (251.7 MB)
// cnt      : [N*6]   u32                       (1.5 MB)
// w_pad    : [KP][C_OUT] f32                   (0.5 MB)
#define WS_CNT_OFF   ((size_t)N_NODES * KP * 4)
#define WS_WPAD_OFF  (WS_CNT_OFF + (size_t)N_NODES * 6 * 4)

// ---- kernel 1: build col_data skeleton: zeros everywhere, xa in slot 6 ----
// grid: (N_NODES, ceil(KP/256)), block 256
__global__ void gc_init_coldata(const float* __restrict__ x,
                                const int* __restrict__ node_type,
                                float* __restrict__ cd) {
    int row = blockIdx.x;
    int c = blockIdx.y * 256 + threadIdx.x;
    if (c >= KP) return;
    float v = 0.0f;
    if (c >= SLOT6 && c < SLOT6 + C_IN) {
        v = x[(size_t)row * C_IN + (c - SLOT6)];
    } else if (c >= SLOT6 + C_IN && c < SLOT6 + FEAT) {
        v = (node_type[row] == (c - SLOT6 - C_IN)) ? 1.0f : 0.0f;
    }
    cd[(size_t)row * KP + c] = v;
}

// ---- kernel 2: zero per-segment counters (slots 0..5 only) ----
__global__ void gc_zero_cnt(unsigned* __restrict__ cnt) {
    int i = blockIdx.x * 256 + threadIdx.x;
    if (i < N_NODES * 6) cnt[i] = 0u;
}

// ---- kernel 3: pad weights [931][128] -> [960][128] (zero tail rows) ----
__global__ void gc_pad_weights(const float* __restrict__ w,
                               float* __restrict__ wp) {
    int i = blockIdx.x * 256 + threadIdx.x;
    if (i >= KP * C_OUT) return;
    wp[i] = (i < K_REAL * C_OUT) ? w[i] : 0.0f;
}

// ---- kernel 4: per-segment edge counts (skip dead edge_type==6) ----
__global__ void gc_count(const int* __restrict__ eidx,
                         const int* __restrict__ etype,
                         unsigned* __restrict__ cnt, int E) {
    int e = blockIdx.x * 256 + threadIdx.x;
    if (e >= E) return;
    int t = etype[e];
    if (t >= 6) return;                // slot 6 overwritten later; dead work
    int row = eidx[e];
    atomicAdd(&cnt[(size_t)row * 6 + t], 1u);
}

// ---- kernel 5: scatter-mean, one wave per edge ----
// xa[col] is read from col_data[col][SLOT6 .. SLOT6+132] (never written by
// this kernel since etype==6 edges are skipped), scaled by 1/max(cnt,1),
// atomically accumulated into col_data[row][t*133 ..].
__global__ void gc_scatter(const int* __restrict__ eidx,
                           const int* __restrict__ etype,
                           const unsigned* __restrict__ cnt,
                           float* __restrict__ cd, int E) {
    int gid = blockIdx.x * 256 + threadIdx.x;
    int e = gid >> 5;
    int lane = gid & 31;
    if (e >= E) return;
    int t = etype[e];
    if (t >= 6) return;                // whole wave skips uniformly
    int row = eidx[e];
    int col = eidx[E + e];
    unsigned c = cnt[(size_t)row * 6 + t];
    float s = 1.0f / (float)(c > 1u ? c : 1u);
    const float* src = cd + (size_t)col * KP + SLOT6;   // xa[col]
    float* dst = cd + (size_t)row * KP + t * FEAT;
    for (int f = lane; f < FEAT; f += 32) {
        unsafeAtomicAdd(dst + f, src[f] * s);
    }
}

// ---- kernel 6: GEMM out[N,128] = col_data[N,KP] @ w_pad[KP,128] ----
// 256 threads = 8 waves; each wave owns a 16x128 output tile via
// V_WMMA_F32_16X16X4_F32 (exact fp32 path). Weights are staged in LDS in
// *B-fragment order*: sF[(kk*8+n)*33 + lane] is the float2 (bv.x, bv.y)
// that lane `lane` of fragment (k-step kk, tile n) needs, so the inner loop
// is exactly one conflict-free ds_load_b64 per WMMA (no repair movs).
__global__ void gc_gemm(const float* __restrict__ A,      // col_data
                        const float* __restrict__ Wp,     // padded weights
                        float* __restrict__ out) {
    __shared__ v2f sF[16 * 8 * 33];                       // 33 KB, 64-row K chunk

    int wave = threadIdx.x >> 5;
    int lane = threadIdx.x & 31;
    int m0 = blockIdx.x * 128 + wave * 16;                // wave's row block

    // A fragment (32-bit A 16x4): lane&15 = M row; lane>>4 selects K pair
    int rowA = m0 + (lane & 15);
    int kofs = (lane >> 4) * 2;
    const float* pA = A + (size_t)rowA * KP + kofs;

    int coln = lane & 15;                                  // output column in tile
    v8f acc[8] = {};

    for (int kb = 0; kb < KP; kb += 64) {
        __syncthreads();
        // Cooperative fill: item enumerates (row-pair kr2, column c).
        // Each item does two coalesced b32 global loads (rows r0, r0+1 of Wp)
        // and one aligned float2 LDS store in fragment order.
        for (int item = threadIdx.x; item < 32 * C_OUT; item += 256) {
            int c    = item & (C_OUT - 1);
            int kr2  = item >> 7;          // 0..31: pair index within chunk
            int kk   = kr2 >> 1;           // k-step 0..15
            int half = kr2 & 1;            // 0 -> lanes 0-15, 1 -> lanes 16-31
            int r0   = kb + kr2 * 2;
            v2f val;
            val.x = Wp[(size_t)r0 * C_OUT + c];
            val.y = Wp[(size_t)(r0 + 1) * C_OUT + c];
            sF[(kk * 8 + (c >> 4)) * 33 + (c & 15) + half * 16] = val;
        }
        __syncthreads();

#pragma unroll 4
        for (int kk = 0; kk < 16; ++kk) {
            v2f av = *(const v2f*)(pA + kb + kk * 4);
#pragma unroll
            for (int n = 0; n < 8; ++n) {
                v2f bv = sF[(kk * 8 + n) * 33 + lane];
                acc[n] = __builtin_amdgcn_wmma_f32_16x16x4_f32(
                    false, av, false, bv, (short)0, acc[n], false, false);
            }
        }
    }

    // C/D layout: VGPR i, lanes 0-15 -> (M=i, N=lane); lanes 16-31 -> (M=8+i)
    int rhalf = (lane >> 4) * 8;
#pragma unroll
    for (int n = 0; n < 8; ++n) {
#pragma unroll
        for (int i = 0; i < 8; ++i) {
            out[(size_t)(m0 + rhalf + i) * C_OUT + n * 16 + coln] = acc[n][i];
        }
    }
}

extern "C" void kernel_launch(void* const* d_in, const int* in_sizes, int n_in,
                              void* d_out, int out_size, void* d_ws, size_t ws_size,
                              hipStream_t stream) {
    const float* x     = (const float*)d_in[0];
    const int*   eidx  = (const int*)d_in[1];   // [2][E]: row then col
    const int*   etype = (const int*)d_in[2];
    const int*   ntype = (const int*)d_in[3];
    const float* w     = (const float*)d_in[4]; // [931][128]
    float* out = (float*)d_out;
    int E = in_sizes[2];

    float*    cd  = (float*)d_ws;
    unsigned* cnt = (unsigned*)((char*)d_ws + WS_CNT_OFF);
    float*    wp  = (float*)((char*)d_ws + WS_WPAD_OFF);

    dim3 g1(N_NODES, (KP + 255) / 256);
    gc_init_coldata<<<g1, 256, 0, stream>>>(x, ntype, cd);
    gc_zero_cnt<<<(N_NODES * 6 + 255) / 256, 256, 0, stream>>>(cnt);
    gc_pad_weights<<<(KP * C_OUT + 255) / 256, 256, 0, stream>>>(w, wp);
    gc_count<<<(E + 255) / 256, 256, 0, stream>>>(eidx, etype, cnt, E);
    int scatter_threads = E * 32;
    gc_scatter<<<(scatter_threads + 255) / 256, 256, 0, stream>>>(eidx, etype, cnt, cd, E);
    gc_gemm<<<N_NODES / 128, 256, 0, stream>>>(cd, wp, out);
}